// SMART_3m_22316650070989
// MI455X (gfx1250) — compile-verified
//
#include <hip/hip_runtime.h>
#include <math.h>

typedef __attribute__((ext_vector_type(2))) float v2f;
typedef __attribute__((ext_vector_type(8))) float v8f;

#define D 128

// ---------------------------------------------------------------- zero
__global__ void k_zero(float* __restrict__ p, size_t n) {
  size_t i = (size_t)blockIdx.x * blockDim.x + threadIdx.x;
  size_t stride = (size_t)gridDim.x * blockDim.x;
  for (; i < n; i += stride) p[i] = 0.0f;
}

// ---------------------------------------------------------------- scatter-add (mean aggregation)
// One wave32 per edge; lane l moves float4 at columns [4l, 4l+3].
// Coalesced global_load_b128 + global_atomic_add_f32 (resolves in L2: agg is 25.6MB << 192MB).
__global__ __launch_bounds__(256)
void k_scatter(const float* __restrict__ x, const int* __restrict__ src,
               const int* __restrict__ dst, float* __restrict__ agg,
               float* __restrict__ cnt, int E) {
  const int lane  = threadIdx.x & 31;
  const int wave  = (int)((blockIdx.x * blockDim.x + threadIdx.x) >> 5);
  const int nwav  = (int)((gridDim.x * blockDim.x) >> 5);
  for (int e = wave; e < E; e += nwav) {
    const int s = src[e];
    const int d = dst[e];
    const float4 v = *(const float4*)(x + (size_t)s * D + lane * 4);
    float* b = agg + (size_t)d * D + lane * 4;
    atomicAdd(b + 0, v.x);
    atomicAdd(b + 1, v.y);
    atomicAdd(b + 2, v.z);
    atomicAdd(b + 3, v.w);
    if (lane == 0) atomicAdd(cnt + d, 1.0f);
  }
}

// ---------------------------------------------------------------- SAGE layer GEMM + bias + L2 norm
// out = (agg/max(cnt,1)) @ Wl^T + b + x @ Wr^T ; rows L2-normalized.
// Block = 8 waves = one 16-row stripe; wave w computes 16x16 tile at cols [16w,16w+16).
__global__ __launch_bounds__(256)
void k_sage_gemm(const float* __restrict__ agg, const float* __restrict__ cnt,
                 const float* __restrict__ x, const float* __restrict__ Wl,
                 const float* __restrict__ bias, const float* __restrict__ Wr,
                 float* __restrict__ out, int nrows) {
  __shared__ float rowsum[16];
  const int lane = threadIdx.x & 31;
  const int wave = threadIdx.x >> 5;    // col tile 0..7
  const int half = lane >> 4;           // 0|1
  const int l16  = lane & 15;
  const size_t row0 = (size_t)blockIdx.x * 16;
  const int col0 = wave * 16;

  if (threadIdx.x < 16) rowsum[threadIdx.x] = 0.0f;
  __syncthreads();

  // A-matrix row handled by this lane (same row for both lane halves)
  size_t arow = row0 + l16;
  if (arow >= (size_t)nrows) arow = (size_t)nrows - 1;   // safe clamp (N%16==0 in practice)
  const float ic = 1.0f / fmaxf(cnt[arow], 1.0f);

  const float* aA = agg + arow * D;
  const float* aX = x   + arow * D;
  // B[k][j] = W[j*D + k]; this lane supplies column j = col0+l16
  const float* wl = Wl + (size_t)(col0 + l16) * D;
  const float* wr = Wr + (size_t)(col0 + l16) * D;

  v8f c = {};
  #pragma unroll 4
  for (int kk = 0; kk < D; kk += 4) {
    const int k2 = kk + 2 * half;
    v2f a, b;
    a.x = aA[k2] * ic;  a.y = aA[k2 + 1] * ic;
    b.x = wl[k2];       b.y = wl[k2 + 1];
    c = __builtin_amdgcn_wmma_f32_16x16x4_f32(false, a, false, b, (short)0, c, false, false);
    a.x = aX[k2];       a.y = aX[k2 + 1];
    b.x = wr[k2];       b.y = wr[k2 + 1];
    c = __builtin_amdgcn_wmma_f32_16x16x4_f32(false, a, false, b, (short)0, c, false, false);
  }

  // bias: every element this lane holds is column col0+l16
  const float bj = bias[col0 + l16];
  float sq[8];
  #pragma unroll
  for (int v = 0; v < 8; ++v) { c[v] += bj; sq[v] = c[v] * c[v]; }

  // reduce squares over the 16 columns of this tile (within each 16-lane half)
  #pragma unroll
  for (int m = 8; m >= 1; m >>= 1)
    #pragma unroll
    for (int v = 0; v < 8; ++v) sq[v] += __shfl_xor(sq[v], m, 32);

  if (l16 == 0) {
    #pragma unroll
    for (int v = 0; v < 8; ++v) atomicAdd(&rowsum[v + 8 * half], sq[v]);
  }
  __syncthreads();

  const int col = col0 + l16;
  #pragma unroll
  for (int v = 0; v < 8; ++v) {
    const size_t r = row0 + v + 8 * half;
    if (r < (size_t)nrows) {
      const float inv = 1.0f / fmaxf(sqrtf(rowsum[v + 8 * half]), 1e-12f);
      out[r * D + col] = c[v] * inv;
    }
  }
}

// ---------------------------------------------------------------- fusion FC: concat(x1,x2,x3) @ Wf^T + bf
__global__ __launch_bounds__(256)
void k_fuse_gemm(const float* __restrict__ x1, const float* __restrict__ x2,
                 const float* __restrict__ x3, const float* __restrict__ Wf,
                 const float* __restrict__ bf, float* __restrict__ out, int nrows) {
  const int lane = threadIdx.x & 31;
  const int wave = threadIdx.x >> 5;
  const int half = lane >> 4;
  const int l16  = lane & 15;
  const size_t row0 = (size_t)blockIdx.x * 16;
  const int col0 = wave * 16;

  size_t arow = row0 + l16;
  if (arow >= (size_t)nrows) arow = (size_t)nrows - 1;

  const float* xs[3] = { x1 + arow * D, x2 + arow * D, x3 + arow * D };
  const float* wrow = Wf + (size_t)(col0 + l16) * (3 * D);

  v8f c = {};
  for (int s = 0; s < 3; ++s) {
    const float* aP = xs[s];
    const float* wP = wrow + s * D;
    #pragma unroll 4
    for (int kk = 0; kk < D; kk += 4) {
      const int k2 = kk + 2 * half;
      v2f a, b;
      a.x = aP[k2];  a.y = aP[k2 + 1];
      b.x = wP[k2];  b.y = wP[k2 + 1];
      c = __builtin_amdgcn_wmma_f32_16x16x4_f32(false, a, false, b, (short)0, c, false, false);
    }
  }

  const float bj = bf[col0 + l16];
  const int col = col0 + l16;
  #pragma unroll
  for (int v = 0; v < 8; ++v) {
    const size_t r = row0 + v + 8 * half;
    if (r < (size_t)nrows) out[r * D + col] = c[v] + bj;
  }
}

// ---------------------------------------------------------------- host orchestration
extern "C" void kernel_launch(void* const* d_in, const int* in_sizes, int n_in,
                              void* d_out, int out_size, void* d_ws, size_t ws_size,
                              hipStream_t stream) {
  const float* f1 = (const float*)d_in[0];
  const float* f2 = (const float*)d_in[1];
  const float* f3 = (const float*)d_in[2];
  const int*  ei1 = (const int*)d_in[3];
  const int*  ei2 = (const int*)d_in[4];
  const int*  ei3 = (const int*)d_in[5];
  const int E1 = in_sizes[3] / 2, E2 = in_sizes[4] / 2, E3 = in_sizes[5] / 2;
  const int NR = in_sizes[0] / D;            // 50000
  const size_t ND = (size_t)NR * D;

  float* ws  = (float*)d_ws;
  float* agg = ws;                           // [N*D]
  float* cnt = ws + ND;                      // [N]
  float* h   = ws + ND + NR;                 // [N*D] temp
  float* outp = (float*)d_out;
  float* xf = outp;                          // fused embedding
  float* b1 = outp + ND;                     // x1 (enc) then x1_rec
  float* b2 = outp + 2 * ND;                 // x2 then x2_rec
  float* b3 = outp + 3 * ND;                 // x3 then x3_rec

  const int tiles = (NR + 15) / 16;          // 3125

  auto sage = [&](const float* xin, const int* ei, int E, int pbase, int conv, float* o) {
    const float* Wl = (const float*)d_in[pbase + conv * 3 + 0];
    const float* bb = (const float*)d_in[pbase + conv * 3 + 1];
    const float* Wr = (const float*)d_in[pbase + conv * 3 + 2];
    k_zero<<<2048, 256, 0, stream>>>(agg, ND + (size_t)NR);           // agg+cnt contiguous
    k_scatter<<<1024, 256, 0, stream>>>(xin, ei, ei + E, agg, cnt, E);
    k_sage_gemm<<<tiles, 256, 0, stream>>>(agg, cnt, xin, Wl, bb, Wr, o, NR);
  };

  // encoders (params at d_in[6..11], [12..17], [18..23])
  sage(f1, ei1, E1, 6, 0, h);  sage(h, ei1, E1, 6, 1, b1);
  sage(f2, ei2, E2, 12, 0, h); sage(h, ei2, E2, 12, 1, b2);
  sage(f3, ei3, E3, 18, 0, h); sage(h, ei3, E3, 18, 1, b3);

  // fusion (Wf=d_in[24], bf=d_in[25])
  k_fuse_gemm<<<tiles, 256, 0, stream>>>(b1, b2, b3,
      (const float*)d_in[24], (const float*)d_in[25], xf, NR);

  // decoders (params at d_in[26..31], [32..37], [38..43]); reconstructions overwrite b1..b3
  sage(xf, ei1, E1, 26, 0, h); sage(h, ei1, E1, 26, 1, b1);
  sage(xf, ei2, E2, 32, 0, h); sage(h, ei2, E2, 32, 1, b2);
  sage(xf, ei3, E3, 38, 0, h); sage(h, ei3, E3, 38, 1, b3);
}